// OnlineDPCClus_4956392259735
// MI455X (gfx1250) — compile-verified
//
#include <hip/hip_runtime.h>
#include <hip/hip_bf16.h>

#define FLT_BIG 3.402823466e+38f

typedef __attribute__((ext_vector_type(2))) float v2f;
typedef __attribute__((ext_vector_type(8))) float v8f;

constexpr int B_    = 8;
constexpr int C_    = 256;
constexpr int HW_   = 4096;        // 64*64
constexpr int ROWS_ = B_ * HW_;    // 32768
constexpr int M_    = 4096;
constexpr int KNN_  = 20;
constexpr int K_    = 8;
constexpr float TEMP_   = 0.1f;
constexpr float BN_EPS_ = 1e-5f;

// ---------------------------------------------------------------------------
// Kernel 1: 1x1 conv as GEMM via V_WMMA_F32_16X16X4_F32.
// P[row, o] = sum_i W[o,i] * X[b, i, n],  row = b*4096 + n.
// A(m,k) = X[b, k, n0+m]; B(k,o) = W[o0+o, k]; D row-striped per ISA layout.
// ---------------------------------------------------------------------------
__global__ __launch_bounds__(256) void k_proj(const float* __restrict__ X,
                                              const float* __restrict__ W,
                                              float* __restrict__ P) {
  const int lane = threadIdx.x & 31;
  const int wave = threadIdx.x >> 5;
  const int tile = blockIdx.x * 8 + wave;   // 2048 rowTiles * 16 colTiles
  const int rowTile = tile >> 4;
  const int colTile = tile & 15;
  const int row0 = rowTile * 16;
  const int b  = row0 >> 12;                // 4096 rows per batch
  const int n0 = row0 & (HW_ - 1);
  const int o0 = colTile * 16;
  const int m  = lane & 15;                 // A row index / B col index
  const int kh = (lane >> 4) * 2;           // K sub-offset for this half-wave

  const float* Xb = X + (size_t)b * C_ * HW_;
  v8f acc = {};
  for (int k0 = 0; k0 < C_; k0 += 4) {
    v2f a, bb;
    a.x = Xb[(size_t)(k0 + kh) * HW_ + n0 + m];
    a.y = Xb[(size_t)(k0 + kh + 1) * HW_ + n0 + m];
    bb  = *(const v2f*)&W[(size_t)(o0 + m) * C_ + k0 + kh];
    acc = __builtin_amdgcn_wmma_f32_16x16x4_f32(false, a, false, bb,
                                                (short)0, acc, false, false);
  }
  const int rbase = row0 + (lane >> 4) * 8;
#pragma unroll
  for (int r = 0; r < 8; ++r)
    P[(size_t)(rbase + r) * C_ + o0 + m] = acc[r];
}

// ---------------------------------------------------------------------------
// Kernels 2a/2b: deterministic per-channel sum / sumsq for BatchNorm.
// ---------------------------------------------------------------------------
__global__ __launch_bounds__(256) void k_chpart(const float* __restrict__ F,
                                                float* __restrict__ part) {
  const int g = blockIdx.x, t = threadIdx.x;
  float s = 0.f, s2 = 0.f;
  for (int r = 0; r < 256; ++r) {
    float v = F[((size_t)g * 256 + r) * C_ + t];
    s += v; s2 += v * v;
  }
  part[g * 512 + t]       = s;
  part[g * 512 + 256 + t] = s2;
}

__global__ __launch_bounds__(256) void k_chfin(const float* __restrict__ part,
                                               float* __restrict__ chsum,
                                               float* __restrict__ chsq) {
  const int t = threadIdx.x;
  float s = 0.f, s2 = 0.f;
  for (int g = 0; g < 128; ++g) {
    s  += part[g * 512 + t];
    s2 += part[g * 512 + 256 + t];
  }
  chsum[t] = s; chsq[t] = s2;
}

// ---------------------------------------------------------------------------
// Kernel 3: BN (training stats) + ReLU in place; per-row squared norm.
// ---------------------------------------------------------------------------
__global__ __launch_bounds__(256) void k_bn(float* __restrict__ F,
                                            const float* __restrict__ chsum,
                                            const float* __restrict__ chsq,
                                            const float* __restrict__ gamma,
                                            const float* __restrict__ beta,
                                            float* __restrict__ fnorm) {
  __shared__ float red[256];
  const int row = blockIdx.x;
  const int o = threadIdx.x;
  const float inv_n = 1.0f / (float)ROWS_;
  float mean = chsum[o] * inv_n;
  float var  = chsq[o] * inv_n - mean * mean;
  float rstd = rsqrtf(var + BN_EPS_);
  float p = F[(size_t)row * C_ + o];
  float v = gamma[o] * (p - mean) * rstd + beta[o];
  v = fmaxf(v, 0.0f);
  F[(size_t)row * C_ + o] = v;
  red[o] = v * v;
  __syncthreads();
  for (int s = 128; s > 0; s >>= 1) {
    if (o < s) red[o] += red[o + s];
    __syncthreads();
  }
  if (o == 0) fnorm[row] = red[0];
}

// Kernel 4: memory bank squared norms.
__global__ __launch_bounds__(256) void k_mnorm(const float* __restrict__ MB,
                                               float* __restrict__ mnorm) {
  __shared__ float red[256];
  const int mrow = blockIdx.x, t = threadIdx.x;
  float v = MB[(size_t)mrow * C_ + t];
  red[t] = v * v;
  __syncthreads();
  for (int s = 128; s > 0; s >>= 1) {
    if (t < s) red[t] += red[t + s];
    __syncthreads();
  }
  if (t == 0) mnorm[mrow] = red[0];
}

// ---------------------------------------------------------------------------
// Kernel 5 (hot): per 16-row tile, full 4096-col d^2 stripe in LDS via WMMA,
// then fused 20-NN selection + adaptive-bandwidth density.
// Dynamic LDS: sA = 16x260 f32 (padded, conflict-free), sD = 16x4096 f32.
// f tile is staged memory->LDS with the CDNA5 async engine (no VGPR bounce),
// drained via s_wait_asynccnt before use.
// ---------------------------------------------------------------------------
__global__ __launch_bounds__(256) void k_dist(const float* __restrict__ F,
                                              const float* __restrict__ MB,
                                              const float* __restrict__ fnorm,
                                              const float* __restrict__ mnorm,
                                              float* __restrict__ density) {
  extern __shared__ float smem[];
  constexpr int SA_STR = 260;                 // 260 % 64 == 4 -> no bank conflicts
  float* sA = smem;                           // 16 x 260
  float* sD = smem + 16 * SA_STR;             // 16 x 4096
  const int tid  = threadIdx.x;
  const int lane = tid & 31;
  const int wave = tid >> 5;
  const int row0 = blockIdx.x * 16;

  // Stage the 16 x 256 f tile into LDS with async b128 copies:
  // 1024 transfers of 16B, 4 per thread. Both sides 16B aligned
  // (LDS row pitch 260 f32 = 1040 B = 65*16 B).
#pragma unroll
  for (int i = 0; i < 4; ++i) {
    const int idx = tid + i * 256;            // 0..1023
    const int r = idx >> 6;                   // 64 b128 chunks per row
    const int c = (idx & 63) * 4;
    const float* gsrc = &F[(size_t)(row0 + r) * C_ + c];
    uint32_t ldst = (uint32_t)(uintptr_t)&sA[r * SA_STR + c];
    asm volatile("global_load_async_to_lds_b128 %0, %1, off"
                 :: "v"(ldst), "v"(gsrc) : "memory");
  }
  asm volatile("s_wait_asynccnt 0x0" ::: "memory");
  __syncthreads();

  const int m  = lane & 15;
  const int kh = (lane >> 4) * 2;
  float fnr[8];
#pragma unroll
  for (int r = 0; r < 8; ++r)
    fnr[r] = fnorm[row0 + r + (lane >> 4) * 8];

  // GEMM phase: each wave sweeps 32 of the 256 column tiles.
  for (int ct = wave; ct < M_ / 16; ct += 8) {
    const int col0 = ct * 16;
    const float mn = mnorm[col0 + m];
    const float* bp = MB + (size_t)(col0 + m) * C_ + kh;
    if (ct + 8 < M_ / 16)
      __builtin_prefetch(MB + (size_t)(col0 + 128 + m) * C_, 0, 1);
    v8f acc = {};
    for (int k0 = 0; k0 < C_; k0 += 4) {
      v2f a  = *(const v2f*)&sA[m * SA_STR + k0 + kh];
      v2f bb = *(const v2f*)&bp[k0];
      acc = __builtin_amdgcn_wmma_f32_16x16x4_f32(false, a, false, bb,
                                                  (short)0, acc, false, false);
    }
#pragma unroll
    for (int r = 0; r < 8; ++r) {
      int rl = r + (lane >> 4) * 8;
      float d2 = fnr[r] + mn - 2.0f * acc[r];
      sD[(size_t)rl * M_ + col0 + m] = fmaxf(d2, 0.0f);
    }
  }
  __syncthreads();

  // Selection + density phase. sA space is dead now: reuse it to stash mins.
  float* stash = sA;                      // 16 rows x KNN_ floats
  for (int rr = wave * 2; rr < wave * 2 + 2; ++rr) {
    float* dr = sD + (size_t)rr * (size_t)M_;
    float lastv = 0.f;
    for (int it = 0; it < KNN_; ++it) {
      float bv = FLT_BIG; int bi = M_;
      for (int j = lane; j < M_; j += 32) {
        float v = dr[j];
        if (v < bv) { bv = v; bi = j; }
      }
#pragma unroll
      for (int off = 16; off > 0; off >>= 1) {
        float ov = __shfl_xor(bv, off, 32);
        int   oi = __shfl_xor(bi, off, 32);
        if (ov < bv || (ov == bv && oi < bi)) { bv = ov; bi = oi; }
      }
      if (lane == 0) { stash[rr * KNN_ + it] = bv; dr[bi] = FLT_BIG; }
      __threadfence_block();
      lastv = bv;
    }
    float rk      = sqrtf(lastv);                 // k-th NN distance
    float bw      = fmaxf(rk, 1e-8f);
    float inv_bw2 = 1.0f / (bw * bw);
    float dens = 0.f;
    for (int j = lane; j < M_; j += 32)
      dens += expf(-dr[j] * inv_bw2);             // replaced entries -> exp(-huge)=0
    if (lane < KNN_)
      dens += expf(-stash[rr * KNN_ + lane] * inv_bw2);
#pragma unroll
    for (int off = 16; off > 0; off >>= 1)
      dens += __shfl_xor(dens, off, 32);
    if (lane == 0) density[row0 + rr] = dens;
  }
}

// ---------------------------------------------------------------------------
// Kernel 6: top-8 density peaks of the LAST batch; centers, priors, norms.
// ---------------------------------------------------------------------------
__global__ __launch_bounds__(256) void k_peaks(const float* __restrict__ density,
                                               const float* __restrict__ F,
                                               const float* __restrict__ fnorm,
                                               float* __restrict__ centers,
                                               float* __restrict__ priors,
                                               float* __restrict__ cnorm) {
  __shared__ float sd[M_];
  __shared__ float rv[256];
  __shared__ int   ri[256];
  __shared__ int   topi[K_];
  __shared__ float topv[K_];
  const int t = threadIdx.x;
  for (int j = t; j < M_; j += 256) sd[j] = density[(B_ - 1) * HW_ + j];
  __syncthreads();
  for (int it = 0; it < K_; ++it) {
    float bv = -FLT_BIG; int bi = 0;
    for (int j = t; j < M_; j += 256) {
      float v = sd[j];
      if (v > bv) { bv = v; bi = j; }
    }
    rv[t] = bv; ri[t] = bi;
    __syncthreads();
    if (t == 0) {
      float gb = -FLT_BIG; int gi = 0;
      for (int q = 0; q < 256; ++q)
        if (rv[q] > gb || (rv[q] == gb && ri[q] < gi)) { gb = rv[q]; gi = ri[q]; }
      topv[it] = gb; topi[it] = gi;
      sd[gi] = -FLT_BIG;
    }
    __syncthreads();
  }
  if (t == 0) {
    float s = 0.f;
    for (int k = 0; k < K_; ++k) s += topv[k];
    s += 1e-8f;
    for (int k = 0; k < K_; ++k) {
      priors[k] = topv[k] / s;
      cnorm[k]  = fnorm[(B_ - 1) * HW_ + topi[k]];
    }
  }
  __syncthreads();
  for (int idx = t; idx < K_ * C_; idx += 256) {
    int k = idx >> 8, c = idx & 255;
    centers[idx] = F[(size_t)((B_ - 1) * HW_ + topi[k]) * C_ + c];
  }
}

// ---------------------------------------------------------------------------
// Kernel 7: soft assignment with density priors. One wave per row.
// ---------------------------------------------------------------------------
__global__ __launch_bounds__(256) void k_assign(const float* __restrict__ F,
                                                const float* __restrict__ centers,
                                                const float* __restrict__ priors,
                                                const float* __restrict__ cnorm,
                                                const float* __restrict__ fnorm,
                                                float* __restrict__ out) {
  __shared__ float sc[K_ * C_];
  __shared__ float sp[K_], sn[K_];
  const int t = threadIdx.x;
  for (int j = t; j < K_ * C_; j += 256) sc[j] = centers[j];
  if (t < K_) { sp[t] = priors[t]; sn[t] = cnorm[t]; }
  __syncthreads();
  const int lane = t & 31;
  const int wave = t >> 5;
  const int gw = blockIdx.x * 8 + wave;
  const int nwaves = gridDim.x * 8;
  for (int row = gw; row < ROWS_; row += nwaves) {
    float fv[8];
#pragma unroll
    for (int j = 0; j < 8; ++j) fv[j] = F[(size_t)row * C_ + lane * 8 + j];
    float dot[K_];
#pragma unroll
    for (int k = 0; k < K_; ++k) {
      float s = 0.f;
#pragma unroll
      for (int j = 0; j < 8; ++j) s += fv[j] * sc[k * C_ + lane * 8 + j];
      dot[k] = s;
    }
#pragma unroll
    for (int k = 0; k < K_; ++k)
#pragma unroll
      for (int off = 16; off > 0; off >>= 1)
        dot[k] += __shfl_xor(dot[k], off, 32);
    float fn = fnorm[row];
    float logit[K_];
    float mx = -FLT_BIG;
#pragma unroll
    for (int k = 0; k < K_; ++k) {
      float d2 = fmaxf(fn + sn[k] - 2.0f * dot[k], 0.0f);
      logit[k] = -sqrtf(d2) * (1.0f / TEMP_);
      mx = fmaxf(mx, logit[k]);
    }
    float se = 0.f, acc = 0.f;
#pragma unroll
    for (int k = 0; k < K_; ++k) {
      float e = expf(logit[k] - mx);
      se += e;
      acc += e * sp[k];
    }
    if (lane == 0) out[row] = acc / se;
  }
}

// ---------------------------------------------------------------------------
extern "C" void kernel_launch(void* const* d_in, const int* in_sizes, int n_in,
                              void* d_out, int out_size, void* d_ws, size_t ws_size,
                              hipStream_t stream) {
  const float* features = (const float*)d_in[0];  // [8,256,64,64]
  const float* conv_w   = (const float*)d_in[1];  // [256,256]
  const float* gamma    = (const float*)d_in[2];  // [256]
  const float* beta     = (const float*)d_in[3];  // [256]
  const float* membank  = (const float*)d_in[4];  // [4096,256]
  float* out = (float*)d_out;                     // [8*4096]

  float* ws     = (float*)d_ws;
  float* F      = ws;                             // 32768*256
  float* part   = F + (size_t)ROWS_ * C_;         // 128*512
  float* chsum  = part + 128 * 512;               // 256
  float* chsq   = chsum + 256;                    // 256
  float* fnorm  = chsq + 256;                     // 32768
  float* mnorm  = fnorm + ROWS_;                  // 4096
  float* dens   = mnorm + M_;                     // 32768
  float* cent   = dens + ROWS_;                   // 8*256
  float* priors = cent + K_ * C_;                 // 8
  float* cnorm  = priors + K_;                    // 8

  k_proj<<<4096, 256, 0, stream>>>(features, conv_w, F);
  k_chpart<<<128, 256, 0, stream>>>(F, part);
  k_chfin<<<1, 256, 0, stream>>>(part, chsum, chsq);
  k_bn<<<ROWS_, 256, 0, stream>>>(F, chsum, chsq, gamma, beta, fnorm);
  k_mnorm<<<M_, 256, 0, stream>>>(membank, mnorm);

  const uint32_t lds5 = (uint32_t)((16 * 260 + 16 * M_) * sizeof(float)); // ~272 KB < 320 KB/WGP
  hipFuncSetAttribute((const void*)k_dist,
                      hipFuncAttributeMaxDynamicSharedMemorySize, (int)lds5);
  k_dist<<<ROWS_ / 16, 256, lds5, stream>>>(F, membank, fnorm, mnorm, dens);

  k_peaks<<<1, 256, 0, stream>>>(dens, F, fnorm, cent, priors, cnorm);
  k_assign<<<512, 256, 0, stream>>>(F, cent, priors, cnorm, fnorm, out);
}